// SP_20555713479170
// MI455X (gfx1250) — compile-verified
//
#include <hip/hip_runtime.h>
#include <stdint.h>

// Problem dims
#define Bdim 32
#define Tdim 4096
#define Hdim 768
#define CH   32                 // T-chunks per batch sample
#define TCH  (Tdim / CH)        // 128 rows per chunk
#define ROWS 8                  // rows per TDM stage (8*768*4 = 24 KB)
#define STAGE_ELEMS (ROWS * Hdim)
#define BLK  256                // 8 waves of 32

typedef float        v2f   __attribute__((ext_vector_type(2)));
typedef float        v8f   __attribute__((ext_vector_type(8)));
typedef unsigned int u32x4 __attribute__((ext_vector_type(4)));
typedef int          i32x8 __attribute__((ext_vector_type(8)));
typedef int          i32x4 __attribute__((ext_vector_type(4)));

// ---------------------------------------------------------------------------
// Kernel 1: per-sample valid length = (first index with mask<0)+1, else T
// ---------------------------------------------------------------------------
__global__ __launch_bounds__(BLK) void lengths_kernel(const float* __restrict__ mask,
                                                      int* __restrict__ len) {
  __shared__ int sMin;
  const int b = blockIdx.x;
  if (threadIdx.x == 0) sMin = Tdim;
  __syncthreads();
  int local = Tdim;
  const float* m = mask + (size_t)b * Tdim;
  for (int t = threadIdx.x; t < Tdim; t += BLK) {
    if (m[t] < 0.0f && t < local) local = t;
  }
  atomicMin(&sMin, local);
  __syncthreads();
  if (threadIdx.x == 0) {
    len[b] = (sMin == Tdim) ? Tdim : (sMin + 1);
  }
}

// ---------------------------------------------------------------------------
// TDM descriptor: 2D tile (rows x 768 f32) Global -> LDS
// Group0: count=1 | lds_addr | global_addr | type=2
// Group1: data_size=4B, tensor_dim0=768, tensor_dim1=rows,
//         tile_dim0=768, tile_dim1=rows, dim0_stride=768
// ---------------------------------------------------------------------------
__device__ __forceinline__ void issue_tdm(uint64_t gaddr, uint32_t lds_off, int rows) {
  u32x4 g0;
  g0.x = 1u;                                                // count=1, user desc
  g0.y = lds_off;                                           // LDS byte address
  g0.z = (uint32_t)gaddr;                                   // global_addr[31:0]
  g0.w = ((uint32_t)(gaddr >> 32) & 0x01FFFFFFu) | (2u << 30); // addr[56:32] | type=2

  i32x8 g1;
  g1[0] = (int)(2u << 16);                  // wg_mask=0, data_size=2 (4 bytes)
  g1[1] = (int)((uint32_t)Hdim << 16);      // tensor_dim0[15:0] @ bits[63:48]
  g1[2] = (int)(((uint32_t)rows) << 16);    // tensor_dim0 hi=0; tensor_dim1[15:0]
  g1[3] = (int)((uint32_t)Hdim << 16);      // tensor_dim1 hi=0; tile_dim0=768
  g1[4] = rows;                             // tile_dim1=rows, tile_dim2=0
  g1[5] = Hdim;                             // tensor_dim0_stride lo32 = 768
  g1[6] = 0;                                // stride hi / tensor_dim1_stride lo
  g1[7] = 0;

  i32x4 z4 = {0, 0, 0, 0};                  // groups 2/3 unused (2D tensor)
  i32x8 z8 = {0, 0, 0, 0, 0, 0, 0, 0};      // extra operand (clang-23 6-arg form)
  __builtin_amdgcn_tensor_load_to_lds(g0, g1, z4, z4, z8, 0);
}

// ---------------------------------------------------------------------------
// Kernel 2: per (b, chunk) partial sums Sx / Sxx over valid rows only.
// TDM double-buffered streaming into LDS; each thread owns 3 h columns.
// ---------------------------------------------------------------------------
__global__ __launch_bounds__(BLK) void partial_kernel(const float* __restrict__ feat,
                                                      const int* __restrict__ len,
                                                      float* __restrict__ part) {
  __shared__ float smem[2][STAGE_ELEMS];    // 48 KB

  const int b   = blockIdx.y;
  const int ch  = blockIdx.x;
  const int tid = threadIdx.x;

  const int length = len[b];
  const int t0 = ch * TCH;
  int tend = t0 + TCH; if (tend > length) tend = length;
  const int nrows   = (tend > t0) ? (tend - t0) : 0;       // only valid rows fetched
  const int nstages = (nrows + ROWS - 1) / ROWS;

  float s0 = 0.f, s1 = 0.f, s2 = 0.f, q0 = 0.f, q1 = 0.f, q2 = 0.f;
  const int h0 = tid, h1 = tid + BLK, h2 = tid + 2 * BLK;

  const uint64_t gbase =
      (uint64_t)(uintptr_t)(feat + ((size_t)b * Tdim + (size_t)t0) * Hdim);
  const uint32_t lds0 = (uint32_t)(uintptr_t)&smem[0][0];
  const uint32_t lds1 = (uint32_t)(uintptr_t)&smem[1][0];

  if (nstages > 0 && tid < 32) {            // wave 0 issues DMA (EXEC ignored by TDM)
    int r0 = (nrows < ROWS) ? nrows : ROWS;
    issue_tdm(gbase, lds0, r0);
  }

  for (int i = 0; i < nstages; ++i) {
    int rcnt = nrows - i * ROWS; if (rcnt > ROWS) rcnt = ROWS;

    if (tid < 32) __builtin_amdgcn_s_wait_tensorcnt(0);    // stage i landed in LDS
    __syncthreads();                                       // visible to all waves

    if (tid < 32 && (i + 1) < nstages) {                   // prefetch stage i+1
      int rn = nrows - (i + 1) * ROWS; if (rn > ROWS) rn = ROWS;
      issue_tdm(gbase + (uint64_t)(i + 1) * ROWS * Hdim * 4,
                ((i + 1) & 1) ? lds1 : lds0, rn);
    }

    const float* bp = &smem[i & 1][0];
    if (rcnt == ROWS) {                     // full stage: compile-time unroll,
#pragma unroll                              // 24 ds_load_b32 batched per thread
      for (int r = 0; r < ROWS; ++r) {
        const float* row = bp + r * Hdim;
        float x0 = row[h0]; s0 += x0; q0 = fmaf(x0, x0, q0);
        float x1 = row[h1]; s1 += x1; q1 = fmaf(x1, x1, q1);
        float x2 = row[h2]; s2 += x2; q2 = fmaf(x2, x2, q2);
      }
    } else {                                // tail stage (block-uniform branch)
      for (int r = 0; r < rcnt; ++r) {
        const float* row = bp + r * Hdim;
        float x0 = row[h0]; s0 += x0; q0 = fmaf(x0, x0, q0);
        float x1 = row[h1]; s1 += x1; q1 = fmaf(x1, x1, q1);
        float x2 = row[h2]; s2 += x2; q2 = fmaf(x2, x2, q2);
      }
    }
    __syncthreads();                        // buffer reuse safety
  }

  float* P = part + ((size_t)b * CH + ch) * (2 * Hdim);
  P[h0] = s0; P[h1] = s1; P[h2] = s2;
  P[Hdim + h0] = q0; P[Hdim + h1] = q1; P[Hdim + h2] = q2;
}

// ---------------------------------------------------------------------------
// Kernel 3: reduce 32 chunk-partials per column with V_WMMA_F32_16X16X4_F32
// (A = ones in row 0 -> C row 0 = column sums; invariant to B K-ordering),
// then mean / unbiased std via precomputed reciprocals (no per-tile divides).
// ---------------------------------------------------------------------------
__device__ __forceinline__ v8f wmma_acc(v2f a, v2f b, v8f c) {
  return __builtin_amdgcn_wmma_f32_16x16x4_f32(false, a, false, b, (short)0, c,
                                               false, false);
}

__global__ __launch_bounds__(BLK) void finalize_kernel(const float* __restrict__ part,
                                                       const int* __restrict__ len,
                                                       float* __restrict__ out) {
  const int b    = blockIdx.x;
  const int tid  = threadIdx.x;
  const int wave = tid >> 5;
  const int lane = tid & 31;
  const int n    = lane & 15;               // output column within tile
  const int half = lane >> 4;               // K sub-slot
  const float cnt    = (float)len[b];
  const float rcp_n  = 1.0f / cnt;          // one divide per block,
  const float rcp_n1 = 1.0f / (cnt - 1.0f); // multiplies thereafter

  const float* P = part + (size_t)b * CH * (2 * Hdim);

  v2f A; A.x = (n == 0) ? 1.0f : 0.0f; A.y = A.x;   // A row0 = ones (16x4)

  for (int pair = wave; pair < Hdim / 16; pair += (BLK / 32)) {  // 48 tiles / 8 waves
    const int colS = pair * 16 + n;
    const int colQ = Hdim + colS;
    v8f accS = {}; v8f accQ = {};
    for (int c0 = 0; c0 < CH; c0 += 4) {    // 8 WMMA steps of K=4 each
      v2f Bs, Bq;
      Bs.x = P[(size_t)(c0 + half)     * (2 * Hdim) + colS];
      Bs.y = P[(size_t)(c0 + 2 + half) * (2 * Hdim) + colS];
      Bq.x = P[(size_t)(c0 + half)     * (2 * Hdim) + colQ];
      Bq.y = P[(size_t)(c0 + 2 + half) * (2 * Hdim) + colQ];
      accS = wmma_acc(A, Bs, accS);         // EXEC all-1s here (uniform flow)
      accQ = wmma_acc(A, Bq, accQ);
    }
    if (lane < 16) {                        // C VGPR0, lanes 0-15 = row M=0
      float s = accS[0], q = accQ[0];
      float mean = s * rcp_n;
      float var  = (q - s * s * rcp_n) * rcp_n1;
      out[(size_t)b * (2 * Hdim) + colS] = mean;
      out[(size_t)b * (2 * Hdim) + colQ] = sqrtf(fmaxf(var, 0.0f));
    }
  }
}

// ---------------------------------------------------------------------------
extern "C" void kernel_launch(void* const* d_in, const int* in_sizes, int n_in,
                              void* d_out, int out_size, void* d_ws, size_t ws_size,
                              hipStream_t stream) {
  (void)in_sizes; (void)n_in; (void)out_size; (void)ws_size;
  const float* feat = (const float*)d_in[0];   // [B, T, H] f32
  const float* mask = (const float*)d_in[1];   // [B, T]   f32
  float* out = (float*)d_out;                  // [B, 2H]  f32

  int*   len  = (int*)d_ws;                            // 32 ints
  float* part = (float*)((char*)d_ws + 512);           // [B][CH][2*H] ~ 6.3 MB

  lengths_kernel<<<Bdim, BLK, 0, stream>>>(mask, len);
  partial_kernel<<<dim3(CH, Bdim), BLK, 0, stream>>>(feat, len, part);
  finalize_kernel<<<Bdim, BLK, 0, stream>>>(part, len, out);
}